// GCN_22179211117196
// MI455X (gfx1250) — compile-verified
//
#include <hip/hip_runtime.h>
#include <math.h>

typedef __attribute__((ext_vector_type(2))) float v2f;
typedef __attribute__((ext_vector_type(8))) float v8f;

static constexpr int IN_DIM  = 256;
static constexpr int HID_DIM = 128;
static constexpr int OUT_DIM = 16;

// ---------------- degree / symmetric norm ----------------
__global__ void k_deg_init(float* deg, int n) {
  int i = blockIdx.x * blockDim.x + threadIdx.x;
  if (i < n) deg[i] = 1.0f;  // self-loop contributes 1 to every node
}

__global__ void k_deg_accum(const int* __restrict__ dst, float* deg, int E) {
  int e = blockIdx.x * blockDim.x + threadIdx.x;
  if (e < E) atomicAdd(&deg[dst[e]], 1.0f);
}

__global__ void k_deg_to_dinv(float* deg, int n) {
  int i = blockIdx.x * blockDim.x + threadIdx.x;
  if (i < n) {
    float d = deg[i];
    deg[i] = (d > 0.0f) ? rsqrtf(d) : 0.0f;  // in place: deg -> dinv
  }
}

// ---------------- dense GEMM via V_WMMA_F32_16X16X4_F32 ----------------
// One wave computes a 16x16 tile of C = A[M,K] x B[K,N] (row-major, fp32 exact).
// ISA 7.12.2 layouts:
//   A 16x4:  lanes 0-15 M=lane  {v0=K0, v1=K1};  lanes 16-31 M=lane-16 {v0=K2, v1=K3}
//   B 4x16:  v0 = rows K0 (lanes 0-15) / K2 (lanes 16-31); v1 = K1 / K3; N = lane%16
//   C 16x16: acc[j] -> row (j + 8*(lane>>4)), col lane%16
template <int K_DIM, int N_DIM>
__global__ void k_gemm_wmma_f32(const float* __restrict__ A,
                                const float* __restrict__ B,
                                float* __restrict__ C, int M) {
  const int wave = (int)((blockIdx.x * blockDim.x + threadIdx.x) >> 5);
  const int lane = threadIdx.x & 31;
  constexpr int TN = N_DIM / 16;
  const int rt = wave / TN;
  const int ct = wave - rt * TN;
  if (rt * 16 >= M) return;  // wave-uniform: EXEC stays all-1s for WMMA

  const int lm = lane & 15;
  const int kh = lane >> 4;  // 0 -> K pair {0,1}, 1 -> K pair {2,3}

  const float* arow = A + (size_t)(rt * 16 + lm) * K_DIM + 2 * kh;
  const float* bcol = B + (size_t)(2 * kh) * N_DIM + ct * 16 + lm;

  v8f acc = {};
#pragma unroll 8
  for (int k = 0; k < K_DIM; k += 4) {
    v2f a, b;
    a[0] = arow[k];
    a[1] = arow[k + 1];
    b[0] = bcol[(size_t)k * N_DIM];
    b[1] = bcol[(size_t)(k + 1) * N_DIM];
    acc = __builtin_amdgcn_wmma_f32_16x16x4_f32(false, a, false, b,
                                                (short)0, acc, false, false);
  }

  float* crow = C + (size_t)(rt * 16 + 8 * kh) * N_DIM + ct * 16 + lm;
#pragma unroll
  for (int j = 0; j < 8; ++j) crow[(size_t)j * N_DIM] = acc[j];
}

// ---------------- sparse aggregation ----------------
// out[i,:] = h[i,:] * dinv[i]^2   (self-loop term; also fully initializes out)
template <int F>
__global__ void k_selfloop(const float* __restrict__ h,
                           const float* __restrict__ dinv,
                           float* __restrict__ out, int n) {
  int idx = blockIdx.x * blockDim.x + threadIdx.x;
  if (idx >= n * F) return;
  int i = idx / F;
  float di = dinv[i];
  out[idx] = h[idx] * (di * di);
}

// one wave per edge; lane handles 4 consecutive features (float4 gather)
__global__ void k_scatter_f128(const int* __restrict__ src,
                               const int* __restrict__ dst,
                               const float* __restrict__ dinv,
                               const float* __restrict__ h,
                               float* out, int E) {
  int idx = blockIdx.x * blockDim.x + threadIdx.x;
  int e = idx >> 5;
  if (e >= E) return;
  int lane = idx & 31;
  int s = src[e], d = dst[e];
  float w = dinv[s] * dinv[d];
  const float4 v = *reinterpret_cast<const float4*>(h + (size_t)s * 128 + lane * 4);
  float* o = out + (size_t)d * 128 + lane * 4;
  atomicAdd(o + 0, v.x * w);
  atomicAdd(o + 1, v.y * w);
  atomicAdd(o + 2, v.z * w);
  atomicAdd(o + 3, v.w * w);
}

// thread per (edge, feature) for the narrow F=16 layer
__global__ void k_scatter_f16(const int* __restrict__ src,
                              const int* __restrict__ dst,
                              const float* __restrict__ dinv,
                              const float* __restrict__ h,
                              float* out, int E) {
  int idx = blockIdx.x * blockDim.x + threadIdx.x;
  int e = idx >> 4;
  if (e >= E) return;
  int f = idx & 15;
  int s = src[e], d = dst[e];
  float w = dinv[s] * dinv[d];
  atomicAdd(&out[(size_t)d * 16 + f], h[(size_t)s * 16 + f] * w);
}

__global__ void k_bias_relu(float* h, const float* __restrict__ b, int total) {
  int idx = blockIdx.x * blockDim.x + threadIdx.x;
  if (idx >= total) return;
  h[idx] = fmaxf(h[idx] + b[idx & (HID_DIM - 1)], 0.0f);
}

// fused bias + row-wise log_softmax over 16 columns, in place
__global__ void k_bias_logsoftmax16(float* out, const float* __restrict__ b, int n) {
  int i = blockIdx.x * blockDim.x + threadIdx.x;
  if (i >= n) return;
  float* row = out + (size_t)i * 16;
  float v[16];
  float m = -INFINITY;
#pragma unroll
  for (int f = 0; f < 16; ++f) {
    v[f] = row[f] + b[f];
    m = fmaxf(m, v[f]);
  }
  float s = 0.0f;
#pragma unroll
  for (int f = 0; f < 16; ++f) s += expf(v[f] - m);
  float ls = logf(s);
#pragma unroll
  for (int f = 0; f < 16; ++f) row[f] = v[f] - m - ls;
}

// ---------------- launcher ----------------
extern "C" void kernel_launch(void* const* d_in, const int* in_sizes, int n_in,
                              void* d_out, int out_size, void* d_ws, size_t ws_size,
                              hipStream_t stream) {
  (void)n_in; (void)out_size; (void)ws_size;

  const float* x  = (const float*)d_in[0];
  const int*   ei = (const int*)d_in[1];   // [2, E] row-major: src then dst
  const float* W1 = (const float*)d_in[2];
  const float* b1 = (const float*)d_in[3];
  const float* W2 = (const float*)d_in[4];
  const float* b2 = (const float*)d_in[5];

  const int n = in_sizes[0] / IN_DIM;
  const int E = in_sizes[1] / 2;
  const int* src = ei;
  const int* dst = ei + E;

  // workspace layout (fp32): dinv[n] | h1[n*128] | agg1[n*128] | h2[n*16]
  float* dinv = (float*)d_ws;
  float* h1   = dinv + n;
  float* agg1 = h1 + (size_t)n * HID_DIM;
  float* h2   = agg1 + (size_t)n * HID_DIM;
  float* out  = (float*)d_out;

  const int B = 256;  // 8 waves per block on wave32
  auto cdiv = [](long long a, long long b) { return (int)((a + b - 1) / b); };

  // normalization factors
  k_deg_init<<<cdiv(n, B), B, 0, stream>>>(dinv, n);
  k_deg_accum<<<cdiv(E, B), B, 0, stream>>>(dst, dinv, E);
  k_deg_to_dinv<<<cdiv(n, B), B, 0, stream>>>(dinv, n);

  // layer 1: h1 = X @ W1 (WMMA), aggregate, bias+ReLU
  {
    long long waves = (long long)cdiv(n, 16) * (HID_DIM / 16);
    k_gemm_wmma_f32<IN_DIM, HID_DIM>
        <<<cdiv(waves * 32, B), B, 0, stream>>>(x, W1, h1, n);
  }
  k_selfloop<HID_DIM><<<cdiv((long long)n * HID_DIM, B), B, 0, stream>>>(h1, dinv, agg1, n);
  k_scatter_f128<<<cdiv((long long)E * 32, B), B, 0, stream>>>(src, dst, dinv, h1, agg1, E);
  k_bias_relu<<<cdiv((long long)n * HID_DIM, B), B, 0, stream>>>(agg1, b1, n * HID_DIM);

  // layer 2: h2 = relu(agg1) @ W2 (WMMA), aggregate into d_out, bias+log_softmax
  {
    long long waves = (long long)cdiv(n, 16) * (OUT_DIM / 16);
    k_gemm_wmma_f32<HID_DIM, OUT_DIM>
        <<<cdiv(waves * 32, B), B, 0, stream>>>(agg1, W2, h2, n);
  }
  k_selfloop<OUT_DIM><<<cdiv((long long)n * OUT_DIM, B), B, 0, stream>>>(h2, dinv, out, n);
  k_scatter_f16<<<cdiv((long long)E * 16, B), B, 0, stream>>>(src, dst, dinv, h2, out, E);
  k_bias_logsoftmax16<<<cdiv(n, B), B, 0, stream>>>(out, b2, n);
}